// NCC_1005022347622
// MI455X (gfx1250) — compile-verified
//
#include <hip/hip_runtime.h>

typedef __attribute__((ext_vector_type(2))) float v2f;
typedef __attribute__((ext_vector_type(8))) float v8f;

#define BATCH 2
#define DD 160
#define HH 192
#define WW 160
#define WIN 9
#define TILE 16
#define CHUNK 20
#define NW 4
#define NTHREADS 128
#define TILES_H (HH / TILE)            // 12
#define TILES_W (WW / TILE)            // 10
#define CHUNKS_D (DD / CHUNK)          // 8
#define NWG (BATCH * CHUNKS_D * TILES_H * TILES_W)  // 1920
#define TOTAL_VOX 9830400.0f           // 2*160*192*160

// dynamic LDS layout (floats)
#define RBUF_OFF 0                      // NW*5*24*16 = 7680
#define RING_OFF 7680                   // 16*5*32*8  = 20480
#define RED_OFF  (7680 + 20480)         // 128
#define SMEM_FLOATS (7680 + 20480 + 128)

__global__ __launch_bounds__(NTHREADS) void ncc_main(const float* __restrict__ I,
                                                     const float* __restrict__ J,
                                                     float* __restrict__ partials) {
  extern __shared__ __align__(16) float smem[];
  float* Rbuf = smem + RBUF_OFF;   // [NW][5][24][16] row box-sums
  float* ring = smem + RING_OFF;   // [16][5][32][8]  HW-boxsum slabs (per lane,v)
  float* red  = smem + RED_OFF;    // [128]

  const int tid  = threadIdx.x;
  const int wave = tid >> 5;
  const int lane = tid & 31;

  int wg = blockIdx.x;
  const int tw = wg % TILES_W;  wg /= TILES_W;
  const int th = wg % TILES_H;  wg /= TILES_H;
  const int cd = wg % CHUNKS_D; wg /= CHUNKS_D;
  const int b  = wg;

  const int h0 = th * TILE;
  const int w0 = tw * TILE;
  const int o0 = cd * CHUNK;

  // Constant banded-ones A matrix N(16x24): N[m][k] = (m <= k <= m+8).
  // ISA 16x4 f32 A layout: lane<16 -> M=lane, K pair {0,1}; lane>=16 -> M=lane-16, K pair {2,3}.
  const int am   = lane & 15;
  const int kofs = (lane >> 4) << 1;
  v2f Amat[6];
#pragma unroll
  for (int c = 0; c < 6; ++c) {
    const int k0 = 4 * c + kofs;
    Amat[c].x = (k0     >= am && k0     <= am + 8) ? 1.0f : 0.0f;
    Amat[c].y = (k0 + 1 >= am && k0 + 1 <= am + 8) ? 1.0f : 0.0f;
  }
  // B(4x16) layout mirror: lane<16 -> col=lane, rows {0,1} of chunk; lane>=16 -> rows {2,3}.
  const int bn   = lane & 15;
  const int rsel = (lane >> 4) << 1;

  float accsum = 0.0f;

  for (int s = 0; s < 7; ++s) {
    const int ls   = 4 * s + wave;   // slab local index 0..27
    const int d    = o0 - 4 + ls;    // absolute slice (may be OOB -> zero slab)
    const int slot = ls & 15;

    // ---- stage 1: load 24x24 halo rows, W-direction sliding box sums into LDS ----
    if (lane < 24) {
      const int hg = h0 - 4 + lane;
      const bool hv = (d >= 0) && (d < DD) && (hg >= 0) && (hg < HH);
      const long long rowbase = (((long long)b * DD + d) * HH + hg) * (long long)WW;
      float vi[24], vj[24];
#pragma unroll
      for (int t = 0; t < 6; ++t) {
        const int wb = w0 - 4 + 4 * t;
        float4 fi = make_float4(0.f, 0.f, 0.f, 0.f);
        float4 fj = make_float4(0.f, 0.f, 0.f, 0.f);
        if (hv && wb >= 0 && wb < WW) {   // 4-aligned chunks never straddle [0,160)
          fi = *(const float4*)(I + rowbase + wb);
          fj = *(const float4*)(J + rowbase + wb);
        }
        vi[4*t+0] = fi.x; vi[4*t+1] = fi.y; vi[4*t+2] = fi.z; vi[4*t+3] = fi.w;
        vj[4*t+0] = fj.x; vj[4*t+1] = fj.y; vj[4*t+2] = fj.z; vj[4*t+3] = fj.w;
      }
      if (hv && (d + NW) < DD) {  // prefetch the slab this wave handles next step
        __builtin_prefetch(I + rowbase + (long long)NW * HH * WW, 0, 1);
        __builtin_prefetch(J + rowbase + (long long)NW * HH * WW, 0, 1);
      }
#pragma unroll
      for (int q = 0; q < 5; ++q) {
        float x[24];
#pragma unroll
        for (int j2 = 0; j2 < 24; ++j2) {
          const float a = vi[j2], c2 = vj[j2];
          x[j2] = (q == 0) ? a : (q == 1) ? c2 : (q == 2) ? a * a
                 : (q == 3) ? c2 * c2 : a * c2;
        }
        float run = 0.0f;
#pragma unroll
        for (int k = 0; k < WIN; ++k) run += x[k];
        float r[16];
        r[0] = run;
#pragma unroll
        for (int i = 1; i < 16; ++i) { run += x[i + 8] - x[i - 1]; r[i] = run; }
        float4* rr = (float4*)&Rbuf[((wave * 5 + q) * 24 + lane) * 16];
        rr[0] = make_float4(r[0],  r[1],  r[2],  r[3]);
        rr[1] = make_float4(r[4],  r[5],  r[6],  r[7]);
        rr[2] = make_float4(r[8],  r[9],  r[10], r[11]);
        rr[3] = make_float4(r[12], r[13], r[14], r[15]);
      }
    }
    __builtin_amdgcn_wave_barrier();
    asm volatile("s_wait_dscnt 0" ::: "memory");

    // ---- stage 2: H-direction box sum = banded-ones matmul via V_WMMA_F32_16X16X4_F32 ----
    // Out(16x16) = N(16x24) * R(24x16), K=24 in 6 chunks of 4. EXEC all-ones here.
#pragma unroll
    for (int q = 0; q < 5; ++q) {
      v8f acc = {0.f, 0.f, 0.f, 0.f, 0.f, 0.f, 0.f, 0.f};
      const float* rq = &Rbuf[(wave * 5 + q) * 24 * 16];
#pragma unroll
      for (int c = 0; c < 6; ++c) {
        v2f bb;
        bb.x = rq[(4 * c + rsel    ) * 16 + bn];
        bb.y = rq[(4 * c + rsel + 1) * 16 + bn];
        acc = __builtin_amdgcn_wmma_f32_16x16x4_f32(
            /*neg_a=*/false, Amat[c], /*neg_b=*/false, bb,
            /*c_mod=*/(short)0, acc, /*reuse_a=*/false, /*reuse_b=*/false);
      }
      float4* dst = (float4*)&ring[((slot * 5 + q) * 32 + lane) * 8];
      dst[0] = make_float4(acc[0], acc[1], acc[2], acc[3]);
      dst[1] = make_float4(acc[4], acc[5], acc[6], acc[7]);
    }
    __syncthreads();

    // ---- stage 3: D-direction 9-slab sum + cc for one output slice per wave ----
    if (s >= 2) {
      const int lo = 4 * (s - 2) + wave;   // output slice local index
      if (lo < CHUNK) {
        float T[5][8];
#pragma unroll
        for (int q = 0; q < 5; ++q)
#pragma unroll
          for (int v = 0; v < 8; ++v) T[q][v] = 0.0f;
#pragma unroll
        for (int k = 0; k < WIN; ++k) {
          const int sl = (lo + k) & 15;
#pragma unroll
          for (int q = 0; q < 5; ++q) {
            const float4* src = (const float4*)&ring[((sl * 5 + q) * 32 + lane) * 8];
            const float4 a0 = src[0], a1 = src[1];
            T[q][0] += a0.x; T[q][1] += a0.y; T[q][2] += a0.z; T[q][3] += a0.w;
            T[q][4] += a1.x; T[q][5] += a1.y; T[q][6] += a1.z; T[q][7] += a1.w;
          }
        }
        const float invw = 1.0f / 729.0f;
#pragma unroll
        for (int v = 0; v < 8; ++v) {
          const float si = T[0][v], sj = T[1][v];
          const float sii = T[2][v], sjj = T[3][v], sij = T[4][v];
          const float cross = sij - si * sj * invw;
          const float ivar  = sii - si * si * invw;
          const float jvar  = sjj - sj * sj * invw;
          accsum += (cross * cross) / (ivar * jvar + 1e-5f);
        }
      }
    }
    __syncthreads();
  }

  red[tid] = accsum;
  __syncthreads();
  if (tid == 0) {
    float t = 0.0f;
    for (int i = 0; i < NTHREADS; ++i) t += red[i];
    partials[blockIdx.x] = t;
  }
}

__global__ __launch_bounds__(256) void ncc_reduce(const float* __restrict__ partials,
                                                  int n, float* __restrict__ out) {
  __shared__ float sm[256];
  float t = 0.0f;
  for (int i = threadIdx.x; i < n; i += 256) t += partials[i];
  sm[threadIdx.x] = t;
  __syncthreads();
  for (int st = 128; st > 0; st >>= 1) {
    if (threadIdx.x < st) sm[threadIdx.x] += sm[threadIdx.x + st];
    __syncthreads();
  }
  if (threadIdx.x == 0) out[0] = -sm[0] / TOTAL_VOX;
}

extern "C" void kernel_launch(void* const* d_in, const int* in_sizes, int n_in,
                              void* d_out, int out_size, void* d_ws, size_t ws_size,
                              hipStream_t stream) {
  (void)in_sizes; (void)n_in; (void)out_size; (void)ws_size;
  const float* I = (const float*)d_in[0];   // y_true
  const float* J = (const float*)d_in[1];   // y_pred
  float* out = (float*)d_out;
  float* partials = (float*)d_ws;           // NWG floats of scratch

  ncc_main<<<dim3(NWG), dim3(NTHREADS), SMEM_FLOATS * sizeof(float), stream>>>(I, J, partials);
  ncc_reduce<<<dim3(1), dim3(256), 0, stream>>>(partials, NWG, out);
}